// CustomISTFT_41867341201710
// MI455X (gfx1250) — compile-verified
//
#include <hip/hip_runtime.h>
#include <hip/hip_bf16.h>

// ISTFT on MI455X (gfx1250, wave32):
//   out_real = OLA( window * (A @ B) ) / win_sum    -- A: 1024x1024 fp32 IDFT basis
//   out_imag = OLA( window * (I0[j] + (-1)^n I512[j])/N ) / win_sum
// A@B is 17.2 GFLOP fp32 -> V_WMMA_F32_16X16X4_F32 (exact fp32 matrix pipe).
// LDS holds pair-interleaved tiles ([k/2][col*2+(k&1)]) so every WMMA fragment
// is one aligned ds_load_b64; LDS is double-buffered with async global->LDS
// copies for A (ASYNCcnt) overlapping the WMMA stream.

#define NFFT   1024
#define HOP    256
#define TFR    8192            // frames
#define FHALF  513
#define EDGE   512
#define LOUT   2096896         // (TFR-1)*HOP + NFFT - 2*EDGE
#define TK     32              // K-chunk staged in LDS (16 k-pairs)
#define NCHUNK (NFFT / TK)     // 32

// LDS row strides (floats). 256/128 payload + pad so stride % 64 == 32:
// the two half-wave pair-rows land on disjoint LDS bank halves.
#define ASTRIDE 288
#define BSTRIDE 160

typedef float v2f __attribute__((ext_vector_type(2)));
typedef float v8f __attribute__((ext_vector_type(8)));
typedef int   v4i __attribute__((ext_vector_type(4)));

#define AS1 __attribute__((address_space(1)))
#define AS3 __attribute__((address_space(3)))

#if __has_builtin(__builtin_amdgcn_global_load_async_to_lds_b128)
#define HAVE_ASYNC 1
#else
#define HAVE_ASYNC 0
#endif

static __device__ __forceinline__ void wait_async0() {
#if __has_builtin(__builtin_amdgcn_s_wait_asynccnt)
    __builtin_amdgcn_s_wait_asynccnt(0);
#else
    asm volatile("s_wait_asynccnt 0x0" ::: "memory");
#endif
}

#if HAVE_ASYNC
static __device__ __forceinline__ void async_copy_b128(const float* gsrc, float* ldst) {
    AS1 float* gp = (AS1 float*)gsrc;          // generic -> global addrspace
    AS3 float* lp = (AS3 float*)ldst;          // generic -> LDS addrspace
    __builtin_amdgcn_global_load_async_to_lds_b128(
        (AS1 v4i*)gp, (AS3 v4i*)lp, 0, 0);
}
#endif

// ---------------------------------------------------------------------------
// Kernel 1: IDFT basis in PAIR-INTERLEAVED layout:
//   Ap[(k>>1)*2048 + m*2 + (k&1)] = coef(k, m)
//   k==0          :  1/N
//   1<=k<=511     :  2*cos(2*pi*k*m/N)/N
//   k==512        :  cos(pi*m)/N
//   513<=k<=1023  : -2*sin(2*pi*(k-512)*m/N)/N
// ---------------------------------------------------------------------------
__global__ __launch_bounds__(256) void build_basis(float* __restrict__ Ap) {
    const int idx = blockIdx.x * 256 + threadIdx.x;   // 0 .. 1024*1024-1
    const int k = idx >> 10;
    const int m = idx & 1023;
    const float scale = 1.0f / 1024.0f;
    const float w0 = 6.135923151e-3f;                 // 2*pi/1024
    float v;
    if (k == 0) {
        v = scale;
    } else if (k <= 512) {
        const int p = (k * m) & 1023;
        const float amp = (k == 512) ? scale : 2.0f * scale;
        v = amp * cosf((float)p * w0);
    } else {
        const int p = ((k - 512) * m) & 1023;
        v = -2.0f * scale * sinf((float)p * w0);
    }
    Ap[(size_t)(k >> 1) * 2048 + (m << 1) + (k & 1)] = v;
}

// ---------------------------------------------------------------------------
// Kernel 2: framesT[j][m] = sum_k A[k][m] * B[k][j]
//   B row k: k<=512 -> z0 row k (real half); k>512 -> z1 row (k-512) (imag half)
// Block tile: M=128 (8 waves x 16), N=64. Wave tile: 16(M) x 64(N), 4 v8f acc.
// Double-buffered LDS; A staged via async global->LDS b128, B via regs+scatter.
// Output stored TRANSPOSED (j-major) so the OLA pass reads contiguously.
// ---------------------------------------------------------------------------
__global__ __launch_bounds__(256) void gemm_idft(const float* __restrict__ Ap,
                                                 const float* __restrict__ z0,
                                                 const float* __restrict__ z1,
                                                 float* __restrict__ framesT) {
    __shared__ float Asp[2][16][ASTRIDE];   // [buf][k-pair][m*2 + parity]
    __shared__ float Bsp[2][16][BSTRIDE];   // [buf][k-pair][n*2 + parity]

    const int tid  = threadIdx.x;
    const int wave = tid >> 5;
    const int lane = tid & 31;
    const int hi   = (lane >> 4) & 1;       // half-wave selects pair (K+2,K+3)
    const int Nblk = blockIdx.x * 64;       // frame-column block
    const int Mblk = blockIdx.y * 128;      // sample-row block

    v8f acc[4] = {};
    const int mloc = wave * 16 + (lane & 15);

    // Per-thread staging coordinates.
    // A: 16 pair-rows x 256 floats = 1024 float4 -> 4 per thread.
    const int a_pl[4] = { (tid + 0) >> 6, (tid + 256) >> 6,
                          (tid + 512) >> 6, (tid + 768) >> 6 };
    const int a_c4 = tid & 63;
    // B: 32 k-rows x 64 floats = 512 float4 -> 2 per thread.
    const int b_kl[2] = { tid >> 4, (tid + 256) >> 4 };
    const int b_n4 = tid & 15;

    float4 bReg[2];
#if !HAVE_ASYNC
    float4 aReg[4];
#endif

    auto stageA = [&](int c, int nb) {      // chunk c -> buffer nb
        #pragma unroll
        for (int i = 0; i < 4; ++i) {
            const float* src = Ap + (size_t)(c * 16 + a_pl[i]) * 2048
                                  + Mblk * 2 + a_c4 * 4;
#if HAVE_ASYNC
            async_copy_b128(src, &Asp[nb][a_pl[i]][a_c4 * 4]);
#else
            aReg[i] = *(const float4*)src;
#endif
        }
    };
    auto loadB = [&](int c) {
        #pragma unroll
        for (int i = 0; i < 2; ++i) {
            const int gk = c * TK + b_kl[i];
            const float* src = (gk <= 512) ? (z0 + (size_t)gk * TFR)
                                           : (z1 + (size_t)(gk - 512) * TFR);
            bReg[i] = *(const float4*)(src + Nblk + b_n4 * 4);
        }
    };
    auto storeTiles = [&](int nb) {
        #pragma unroll
        for (int i = 0; i < 2; ++i) {       // interleave-scatter B
            const int pr = b_kl[i] >> 1, e = b_kl[i] & 1, n0 = b_n4 * 4;
            Bsp[nb][pr][(n0 + 0) * 2 + e] = bReg[i].x;
            Bsp[nb][pr][(n0 + 1) * 2 + e] = bReg[i].y;
            Bsp[nb][pr][(n0 + 2) * 2 + e] = bReg[i].z;
            Bsp[nb][pr][(n0 + 3) * 2 + e] = bReg[i].w;
        }
#if !HAVE_ASYNC
        #pragma unroll
        for (int i = 0; i < 4; ++i)
            *(float4*)&Asp[nb][a_pl[i]][a_c4 * 4] = aReg[i];
#endif
    };

    // Prologue: stage chunk 0 into buffer 0.
    stageA(0, 0);
    loadB(0);
    storeTiles(0);
#if HAVE_ASYNC
    wait_async0();
#endif
    __syncthreads();

    int cur = 0;
    #pragma unroll 1
    for (int c = 0; c < NCHUNK; ++c) {
        const int nxt = cur ^ 1;
        if (c + 1 < NCHUNK) {               // prefetch next chunk
            stageA(c + 1, nxt);
            loadB(c + 1);
        }
        // Compute chunk c: 8 k-steps x 4 N-subtiles. Each fragment is one
        // aligned ds_load_b64 thanks to the pair-interleaved layout.
        #pragma unroll
        for (int ks8 = 0; ks8 < 8; ++ks8) {
            const int pp = ks8 * 2 + hi;
            const v2f a = *(const v2f*)&Asp[cur][pp][mloc * 2];
            #pragma unroll
            for (int nt = 0; nt < 4; ++nt) {
                const int nloc = nt * 16 + (lane & 15);
                const v2f b = *(const v2f*)&Bsp[cur][pp][nloc * 2];
                acc[nt] = __builtin_amdgcn_wmma_f32_16x16x4_f32(
                    false, a, false, b, (short)0, acc[nt], false, false);
            }
        }
        if (c + 1 < NCHUNK) {
            storeTiles(nxt);
#if HAVE_ASYNC
            wait_async0();
#endif
        }
        __syncthreads();
        cur = nxt;
    }

    // D layout (16x16 f32): lanes 0-15 hold M=r (VGPR r), lanes 16-31 M=r+8.
    // Store transposed: framesT[j][m]; each lane writes 8 consecutive m.
    const int jrow  = Nblk + (lane & 15);
    const int mbase = Mblk + wave * 16 + hi * 8;
    #pragma unroll
    for (int nt = 0; nt < 4; ++nt) {
        float* dst = framesT + (size_t)(jrow + nt * 16) * NFFT + mbase;
        float4 lo = make_float4(acc[nt][0], acc[nt][1], acc[nt][2], acc[nt][3]);
        float4 hi4 = make_float4(acc[nt][4], acc[nt][5], acc[nt][6], acc[nt][7]);
        *(float4*)(dst)     = lo;
        *(float4*)(dst + 4) = hi4;
    }
}

// ---------------------------------------------------------------------------
// Kernel 3: gather overlap-add + window-sum normalize + edge trim.
// Deterministic (no float atomics). framesT is j-major -> contiguous reads.
// Imag channel built directly from z1 rows 0 and 512.
// ---------------------------------------------------------------------------
__global__ __launch_bounds__(256) void ola(const float* __restrict__ framesT,
                                           const float* __restrict__ z1,
                                           const float* __restrict__ window,
                                           float* __restrict__ out) {
    __shared__ float win[NFFT];
    const int tid = threadIdx.x;
    #pragma unroll
    for (int i = tid; i < NFFT; i += 256) win[i] = window[i];
    __syncthreads();

    const int t = blockIdx.x * 256 + tid;           // 0..LOUT-1 (exact grid)
    if (t >= LOUT) return;
    const int tu = t + EDGE;                        // untrimmed coordinate

    int jlo = tu - (NFFT - 1);
    jlo = (jlo < 0) ? 0 : ((jlo + HOP - 1) >> 8);
    int jhi = tu >> 8;
    if (jhi > TFR - 1) jhi = TFR - 1;

    float sr = 0.0f, si = 0.0f, wsum = 0.0f;
    const float invN = 1.0f / 1024.0f;
    for (int j = jlo; j <= jhi; ++j) {              // <= 4 iterations
        const int n   = tu - (j << 8);              // 0..1023
        const float w = win[n];
        wsum += w;
        sr += w * framesT[(size_t)j * NFFT + n];
        const float i0   = z1[j];                   // z[1][0][j]
        const float i512 = z1[(size_t)512 * TFR + j]; // z[1][512][j]
        si += w * invN * (i0 + ((n & 1) ? -i512 : i512));
    }
    if (wsum >= 1e-6f) { sr /= wsum; si /= wsum; }
    out[t]        = sr;
    out[LOUT + t] = si;
}

// ---------------------------------------------------------------------------
extern "C" void kernel_launch(void* const* d_in, const int* in_sizes, int n_in,
                              void* d_out, int out_size, void* d_ws, size_t ws_size,
                              hipStream_t stream) {
    const float* z      = (const float*)d_in[0];           // [2][513][8192]
    const float* window = (const float*)d_in[1];           // [1024]
    const float* z0 = z;                                   // real half rows
    const float* z1 = z + (size_t)FHALF * TFR;             // imag half rows
    float* out = (float*)d_out;                            // [2][LOUT]

    float* Ap      = (float*)d_ws;                         // 1024*1024 floats (4 MB)
    float* framesT = Ap + (size_t)NFFT * NFFT;             // 8192*1024 floats (32 MB)

    build_basis<<<(NFFT * NFFT) / 256, 256, 0, stream>>>(Ap);

    dim3 ggrid(TFR / 64, NFFT / 128);                      // (128, 8)
    gemm_idft<<<ggrid, 256, 0, stream>>>(Ap, z0, z1, framesT);

    ola<<<LOUT / 256, 256, 0, stream>>>(framesT, z1, window, out);
}